// CausalLayerNorm_45372034515002
// MI455X (gfx1250) — compile-verified
//
#include <hip/hip_runtime.h>

typedef __attribute__((ext_vector_type(2))) float v2f;
typedef __attribute__((ext_vector_type(4))) float v4f;
typedef __attribute__((ext_vector_type(8))) float v8f;

#define CLN_B 8
#define CLN_K 16000
#define CLN_H 256
#define CLN_ROWS (CLN_B * CLN_K)
#define CLN_EPS 1e-8f

// ---------------------------------------------------------------------------
// Kernel 1: per-row Sum(x) and Sum(x^2) over H=256, 16 rows per wave,
// reduced with V_WMMA_F32_16X16X4_F32 against a ones B-matrix.
// A-fragment (16x4 f32): lanes 0-15 hold row M=lane, K={k,k+1};
// lanes 16-31 hold row M=lane-16, K={k+2,k+3}.  -> per-lane float2 loads.
// D layout (16x16 f32): VGPR j = row M=j (lanes 0-15) / M=j+8 (lanes 16-31),
// every N column identical, so lane 0 / lane 16 hold all 16 row sums.
// ---------------------------------------------------------------------------
__global__ __launch_bounds__(256) void cln_rowsums(const float* __restrict__ x,
                                                   float* __restrict__ S1,
                                                   float* __restrict__ S2) {
  const int lane = threadIdx.x & 31;
  const int wave = threadIdx.x >> 5;
  const int rowBase = blockIdx.x * 128 + wave * 16;  // 16 rows per wave
  const float* p = x + (size_t)rowBase * CLN_H
                     + (size_t)(lane & 15) * CLN_H
                     + (size_t)((lane >> 4) << 1);

  v2f ones; ones[0] = 1.0f; ones[1] = 1.0f;
  v8f cs = {};  // accumulates Sum(x)
  v8f cq = {};  // accumulates Sum(x*x)

#pragma unroll 8
  for (int k = 0; k < CLN_H; k += 4) {
    v2f a = *(const v2f*)(p + k);
    v2f a2; a2[0] = a[0] * a[0]; a2[1] = a[1] * a[1];
    cs = __builtin_amdgcn_wmma_f32_16x16x4_f32(false, a,  false, ones,
                                               (short)0, cs, false, false);
    cq = __builtin_amdgcn_wmma_f32_16x16x4_f32(false, a2, false, ones,
                                               (short)0, cq, false, false);
  }

  if (lane == 0) {
#pragma unroll
    for (int j = 0; j < 8; ++j) {
      S1[rowBase + j] = cs[j];
      S2[rowBase + j] = cq[j];
    }
  } else if (lane == 16) {
#pragma unroll
    for (int j = 0; j < 8; ++j) {
      S1[rowBase + 8 + j] = cs[j];
      S2[rowBase + 8 + j] = cq[j];
    }
  }
}

// ---------------------------------------------------------------------------
// Kernel 2: in-place inclusive prefix sum over t (length K) per batch.
// One block per batch; chunked block scan: wave32 shuffle scan + LDS combine.
// ---------------------------------------------------------------------------
__global__ __launch_bounds__(256) void cln_scan(float* __restrict__ S1,
                                                float* __restrict__ S2) {
  const int b = blockIdx.x;
  const int tid = threadIdx.x;
  const int lane = tid & 31;
  const int wave = tid >> 5;
  __shared__ float w1[8];
  __shared__ float w2[8];

  float c1 = 0.0f, c2 = 0.0f;  // running carry across chunks
  const int base = b * CLN_K;

  for (int chunk = 0; chunk < CLN_K; chunk += 256) {
    const int t = chunk + tid;
    const bool valid = (t < CLN_K);
    float s1 = valid ? S1[base + t] : 0.0f;
    float s2 = valid ? S2[base + t] : 0.0f;

    // wave32 inclusive scan
#pragma unroll
    for (int d = 1; d < 32; d <<= 1) {
      float o1 = __shfl_up(s1, d, 32);
      float o2 = __shfl_up(s2, d, 32);
      if (lane >= d) { s1 += o1; s2 += o2; }
    }
    if (lane == 31) { w1[wave] = s1; w2[wave] = s2; }
    __syncthreads();

    float off1 = 0.0f, off2 = 0.0f, tot1 = 0.0f, tot2 = 0.0f;
#pragma unroll
    for (int w = 0; w < 8; ++w) {
      const float a1 = w1[w], a2 = w2[w];
      if (w < wave) { off1 += a1; off2 += a2; }
      tot1 += a1; tot2 += a2;
    }
    s1 += off1 + c1;
    s2 += off2 + c2;
    if (valid) { S1[base + t] = s1; S2[base + t] = s2; }
    c1 += tot1;
    c2 += tot2;
    __syncthreads();
  }
}

// ---------------------------------------------------------------------------
// Kernel 3: apply normalization. One wave per row (8 floats per lane).
// Output written with non-temporal stores so the streamed output does not
// evict the (L2-resident, 131 MB < 192 MB) input between passes.
// ---------------------------------------------------------------------------
__global__ __launch_bounds__(256) void cln_apply(const float* __restrict__ x,
                                                 const float* __restrict__ gamma,
                                                 const float* __restrict__ beta,
                                                 const float* __restrict__ P1,
                                                 const float* __restrict__ P2,
                                                 float* __restrict__ out) {
  const int lane = threadIdx.x & 31;
  const int wave = threadIdx.x >> 5;
  const int r = blockIdx.x * 8 + wave;   // flat row = b*K + t
  const int t = r % CLN_K;

  const float invH = 1.0f / (float)CLN_H;
  const float count = (float)(t + 1);
  const float csum = P1[r] * invH;
  const float cpow = P2[r] * invH;
  const float mean = csum / count;
  const float var  = (cpow - 2.0f * mean * csum) / count + mean * mean;
  const float rstd = rsqrtf(var + CLN_EPS);

  const size_t off = (size_t)r * CLN_H + (size_t)lane * 8;
  const v4f* xp = (const v4f*)(x + off);
  const v4f* gp = (const v4f*)(gamma + lane * 8);
  const v4f* bp = (const v4f*)(beta + lane * 8);
  v4f* op = (v4f*)(out + off);

#pragma unroll
  for (int i = 0; i < 2; ++i) {
    const v4f v = xp[i];
    const v4f g = gp[i];
    const v4f be = bp[i];
    v4f o;
    o[0] = g[0] * ((v[0] - mean) * rstd) + be[0];
    o[1] = g[1] * ((v[1] - mean) * rstd) + be[1];
    o[2] = g[2] * ((v[2] - mean) * rstd) + be[2];
    o[3] = g[3] * ((v[3] - mean) * rstd) + be[3];
    __builtin_nontemporal_store(o, &op[i]);
  }
}

// ---------------------------------------------------------------------------
extern "C" void kernel_launch(void* const* d_in, const int* in_sizes, int n_in,
                              void* d_out, int out_size, void* d_ws, size_t ws_size,
                              hipStream_t stream) {
  (void)in_sizes; (void)n_in; (void)out_size; (void)ws_size;
  const float* x     = (const float*)d_in[0];
  const float* gamma = (const float*)d_in[1];
  const float* beta  = (const float*)d_in[2];
  float* out = (float*)d_out;

  float* S1 = (float*)d_ws;          // CLN_ROWS floats
  float* S2 = S1 + CLN_ROWS;         // CLN_ROWS floats  (total ~1 MB)

  cln_rowsums<<<CLN_ROWS / 128, 256, 0, stream>>>(x, S1, S2);
  cln_scan<<<CLN_B, 256, 0, stream>>>(S1, S2);
  cln_apply<<<CLN_ROWS / 8, 256, 0, stream>>>(x, gamma, beta, S1, S2, out);
}